// ExteriorDerivative_83519934038690
// MI455X (gfx1250) — compile-verified
//
#include <hip/hip_runtime.h>
#include <math.h>

#define DIMN   7
#define HID    32
#define NOUT   35
#define NPAD   48
#define FD_EPS 0.0001f

typedef __attribute__((ext_vector_type(2))) float v2f;
typedef __attribute__((ext_vector_type(8))) float v8f;

// Folded constant matrix: out[b,:] = cos(x[b]@W1) @ A2   (32 x 48, cols>=35 zero)
__device__ float g_A2[HID * NPAD];

__device__ __forceinline__ int pos2(int a, int b) {  // index of (a,b), a<b, in C(7,2) order
    return a * (13 - a) / 2 + (b - a - 1);
}

// ---------------------------------------------------------------------------
// Setup: build A2 from W1 (7x32) and W2 (32x21).  Runs once per launch, tiny.
// Uses the exact identity  sin(h+d)-sin(h-d) = 2 cos(h) sin(d)  to fold the
// central difference, the W2 projection, and the signed 3-term scatter of the
// exterior derivative into a single 32x35 matrix:
//   A2[j,o] = (1/eps) * sum over triple o=(i,jj,kk), i<jj<kk:
//       + sin(eps*W1[i ,j]) * W2[j, pos2(jj,kk)]
//       - sin(eps*W1[jj,j]) * W2[j, pos2(i ,kk)]
//       + sin(eps*W1[kk,j]) * W2[j, pos2(i ,jj)]
// ---------------------------------------------------------------------------
__global__ void build_A2_kernel(const float* __restrict__ W1,
                                const float* __restrict__ W2) {
    int idx   = blockIdx.x * blockDim.x + threadIdx.x;
    int total = HID * NPAD;
    for (int e = idx; e < total; e += gridDim.x * blockDim.x) {
        int o = e % NPAD;
        int j = e / NPAD;
        float acc = 0.0f;
        if (o < NOUT) {
            int t = 0;
            for (int i = 0; i < DIMN; ++i)
                for (int jj = i + 1; jj < DIMN; ++jj)
                    for (int kk = jj + 1; kk < DIMN; ++kk) {
                        if (t == o) {
                            acc += sinf(FD_EPS * W1[i  * HID + j]) * W2[j * 21 + pos2(jj, kk)];
                            acc -= sinf(FD_EPS * W1[jj * HID + j]) * W2[j * 21 + pos2(i,  kk)];
                            acc += sinf(FD_EPS * W1[kk * HID + j]) * W2[j * 21 + pos2(i,  jj)];
                        }
                        ++t;
                    }
        }
        g_A2[e] = acc * (1.0f / FD_EPS);
    }
}

// ---------------------------------------------------------------------------
// Main kernel: one wave <-> 16 batch rows per tile, ~4 tiles per wave.
//   A (16x32)  = cos(x@W1) fragments, built in-lane (f32 WMMA A layout:
//                lanes 0-15 carry K=4q+{0,1}, lanes 16-31 carry K=4q+{2,3})
//   B (32x48)  = g_A2, hoisted into 48 VGPRs (v2f b[3][8]) once per wave
//   D (16x48)  = three chained f32 16x16x4 accumulator tiles
// ---------------------------------------------------------------------------
__global__ void __launch_bounds__(256)
ext_deriv_wmma_kernel(const float* __restrict__ x,
                      const float* __restrict__ W1,
                      float* __restrict__ out,
                      int nrows) {
    __shared__ float sW1[DIMN * HID];   // 7x32 row-major, broadcast reads

    for (int e = threadIdx.x; e < DIMN * HID; e += blockDim.x)
        sW1[e] = W1[e];
    __syncthreads();

    const int lane    = threadIdx.x & 31;
    const int m       = lane & 15;            // row-in-tile / column-in-tile
    const int boff    = (lane & 16) ? 2 : 0;  // K sub-offset owned by this half-wave
    const int waveId  = blockIdx.x * (blockDim.x >> 5) + (threadIdx.x >> 5);
    const int nwaves  = gridDim.x * (blockDim.x >> 5);
    const int ntiles  = (nrows + 15) >> 4;

    // Hoist B-matrix fragments (reused by every tile this wave touches):
    // b[n][q] covers K = 4q+boff, 4q+boff+1 for output columns 16n + m.
    v2f b[3][8];
#pragma unroll
    for (int n = 0; n < 3; ++n)
#pragma unroll
        for (int q = 0; q < 8; ++q) {
            int col = 16 * n + m;
            b[n][q].x = g_A2[(4 * q + boff + 0) * NPAD + col];
            b[n][q].y = g_A2[(4 * q + boff + 1) * NPAD + col];
        }

    for (int tile = waveId; tile < ntiles; tile += nwaves) {
        int row  = tile * 16 + m;
        int rsrc = (row < nrows) ? row : (nrows - 1);   // clamp: keep EXEC full for WMMA

        // Prefetch next tile's x rows into cache (global_prefetch_b8).
        {
            int tnext = tile + nwaves;
            if (tnext < ntiles) {
                int rpf = tnext * 16 + m;
                if (rpf < nrows)
                    __builtin_prefetch(&x[(size_t)rpf * DIMN], 0, 1);
            }
        }

        float xv[DIMN];
#pragma unroll
        for (int c = 0; c < DIMN; ++c)
            xv[c] = x[(size_t)rsrc * DIMN + c];

        // cos(h_j) for the 16 hidden indices this lane feeds into A fragments.
        // Native v_cos_f32: abs err ~1e-6, far below the reference FD's own
        // ~1e-3 cancellation noise.
        float ch[16];
#pragma unroll
        for (int q = 0; q < 8; ++q) {
            int j0 = 4 * q + boff;
            float a0 = 0.0f, a1 = 0.0f;
#pragma unroll
            for (int c = 0; c < DIMN; ++c) {
                a0 = fmaf(xv[c], sW1[c * HID + j0],     a0);
                a1 = fmaf(xv[c], sW1[c * HID + j0 + 1], a1);
            }
            ch[2 * q]     = __cosf(a0);
            ch[2 * q + 1] = __cosf(a1);
        }

        v8f acc0 = {}, acc1 = {}, acc2 = {};
#pragma unroll
        for (int q = 0; q < 8; ++q) {
            v2f a;
            a.x = ch[2 * q];
            a.y = ch[2 * q + 1];
            acc0 = __builtin_amdgcn_wmma_f32_16x16x4_f32(false, a, false, b[0][q],
                                                         (short)0, acc0, false, false);
            acc1 = __builtin_amdgcn_wmma_f32_16x16x4_f32(false, a, false, b[1][q],
                                                         (short)0, acc1, false, false);
            acc2 = __builtin_amdgcn_wmma_f32_16x16x4_f32(false, a, false, b[2][q],
                                                         (short)0, acc2, false, false);
        }

        // D layout: VGPR v -> row (v | v+8), column = 16n + m.  Each VGPR's
        // 16-lane half writes 64B contiguous per row.
        const int rhalf = (lane & 16) ? 8 : 0;
#pragma unroll
        for (int v = 0; v < 8; ++v) {
            int r = tile * 16 + v + rhalf;
            if (r < nrows) {
                out[(size_t)r * NOUT + m]      = acc0[v];
                out[(size_t)r * NOUT + 16 + m] = acc1[v];
                if (m < 3)
                    out[(size_t)r * NOUT + 32 + m] = acc2[v];
            }
        }
    }
}

// ---------------------------------------------------------------------------
extern "C" void kernel_launch(void* const* d_in, const int* in_sizes, int n_in,
                              void* d_out, int out_size, void* d_ws, size_t ws_size,
                              hipStream_t stream) {
    (void)n_in; (void)out_size; (void)d_ws; (void)ws_size;

    const float* x  = (const float*)d_in[0];   // (B, 7)
    const float* W1 = (const float*)d_in[1];   // (7, 32)
    const float* W2 = (const float*)d_in[2];   // (32, 21)
    float* out      = (float*)d_out;           // (B, 35)

    int nrows = in_sizes[0] / DIMN;
    if (nrows <= 0) return;

    build_A2_kernel<<<6, 256, 0, stream>>>(W1, W2);

    // ~4 tiles per wave: amortizes the per-wave B-fragment hoist / LDS fill
    // while keeping ~4K waves in flight for latency hiding.
    int ntiles = (nrows + 15) / 16;            // one wave-tile = 16 rows
    int wavesPerBlock = 256 / 32;
    int tilesPerWave  = 4;
    int blocks = (ntiles + wavesPerBlock * tilesPerWave - 1) / (wavesPerBlock * tilesPerWave);
    if (blocks < 1) blocks = 1;
    ext_deriv_wmma_kernel<<<blocks, 256, 0, stream>>>(x, W1, out, nrows);
}